// GroundedCoreferencer_67482526155112
// MI455X (gfx1250) — compile-verified
//
#include <hip/hip_runtime.h>
#include <math.h>
#include <stdint.h>

// Sizes fixed by the reference
#define B_  64
#define NS  512
#define NR  256
#define DD  1024
#define NEGV (-1e10f)

typedef __attribute__((ext_vector_type(2))) float v2f;
typedef __attribute__((ext_vector_type(8))) float v8f;

#if defined(__gfx1250__) && __has_builtin(__builtin_amdgcn_s_wait_asynccnt)
#define HAS_ASYNC_LDS 1
typedef __attribute__((address_space(3))) void as3_void;
#else
#define HAS_ASYNC_LDS 0
#endif

__device__ __forceinline__ v8f zero8() {
    v8f z = {0.f, 0.f, 0.f, 0.f, 0.f, 0.f, 0.f, 0.f};
    return z;
}

__device__ __forceinline__ v8f wmma_f32(v2f a, v2f b, v8f c) {
    // D = A(16x4 f32) * B(4x16 f32) + C(16x16 f32)
    return __builtin_amdgcn_wmma_f32_16x16x4_f32(
        /*neg_a=*/false, a, /*neg_b=*/false, b,
        /*c_mod=*/(short)0, c, /*reuse_a=*/false, /*reuse_b=*/false);
}

// 16-byte global -> LDS copy; async (ASYNCcnt-tracked, bypasses VGPRs) on gfx1250.
__device__ __forceinline__ void copy16_g2l(const float* g, float* l) {
#if HAS_ASYNC_LDS
    // VDST = 32-bit LDS byte offset, VADDR = 64-bit global address (GV mode).
    const unsigned long long ga = (unsigned long long)(uintptr_t)g;
    const unsigned int       lo = (unsigned int)(uintptr_t)(as3_void*)l;
    asm volatile("global_load_async_to_lds_b128 %0, %1, off"
                 :: "v"(lo), "v"(ga) : "memory");
#else
    *(float4*)l = *(const float4*)g;
#endif
}

__device__ __forceinline__ void copy_wait() {
#if HAS_ASYNC_LDS
    __builtin_amdgcn_s_wait_asynccnt(0);
#endif
}

// ---------------------------------------------------------------------------
// Kernel 0: zero the score accumulator
// ---------------------------------------------------------------------------
__global__ void init_kernel(float* score) {
    if (threadIdx.x == 0) *score = 0.f;
}

// ---------------------------------------------------------------------------
// Kernel 1: att[b,s,r] = (span[b,s,:] . image[b,r,:]) * mask ; 0 -> -1e10
// Block = 256 threads (8 waves), computes a 64(M=s) x 64(N=r) tile.
// K=1024 staged through double-buffered LDS in chunks of 32 via async copies.
// ---------------------------------------------------------------------------
__global__ __launch_bounds__(256) void att_kernel(
    const float* __restrict__ span, const float* __restrict__ image,
    const int* __restrict__ smask, const int* __restrict__ imask,
    float* __restrict__ att) {
    // stride 36 floats = 144B: 16B-aligned rows, conflict-free (36m mod 64 distinct)
    __shared__ float As[2][64][36];
    __shared__ float Bs[2][64][36];

    const int b  = blockIdx.z;
    const int s0 = blockIdx.x * 64;
    const int r0 = blockIdx.y * 64;
    const int tid  = threadIdx.x;
    const int wave = tid >> 5;
    const int lane = tid & 31;
    const int lm   = lane & 15;
    const int lk2  = (lane >> 4) << 1;   // 0 or 2 : K sub-pair per half-wave
    const int hh   = lane >> 4;
    const int mw   = wave >> 1;          // 0..3 : M sub-tile (16 rows)
    const int nw   = wave & 1;           // 0..1 : N strip of 32 cols

    const float* spanb = span  + (size_t)b * NS * DD;
    const float* imgb  = image + (size_t)b * NR * DD;

    v8f acc0 = zero8();
    v8f acc1 = zero8();

    // stage: 64 rows x 32 cols each of A and B
    // 512 x 16B copies per matrix / 256 threads = 2 per thread per matrix
    const int srow = tid >> 3;            // 0..31 (+32 on second pass)
    const int sc4  = (tid & 7) * 4;       // 0,4,...,28

    // prologue: fill buffer 0
    {
        copy16_g2l(&spanb[(size_t)(s0 + srow) * DD + sc4],      &As[0][srow][sc4]);
        copy16_g2l(&imgb [(size_t)(r0 + srow) * DD + sc4],      &Bs[0][srow][sc4]);
        copy16_g2l(&spanb[(size_t)(s0 + srow + 32) * DD + sc4], &As[0][srow + 32][sc4]);
        copy16_g2l(&imgb [(size_t)(r0 + srow + 32) * DD + sc4], &Bs[0][srow + 32][sc4]);
    }
    copy_wait();
    __syncthreads();

    for (int chunk = 0; chunk < DD / 32; ++chunk) {
        const int cur = chunk & 1;
        // kick off async fill of the other buffer
        if (chunk + 1 < DD / 32) {
            const int k0 = (chunk + 1) * 32;
            const int nxt = cur ^ 1;
            copy16_g2l(&spanb[(size_t)(s0 + srow) * DD + k0 + sc4],      &As[nxt][srow][sc4]);
            copy16_g2l(&imgb [(size_t)(r0 + srow) * DD + k0 + sc4],      &Bs[nxt][srow][sc4]);
            copy16_g2l(&spanb[(size_t)(s0 + srow + 32) * DD + k0 + sc4], &As[nxt][srow + 32][sc4]);
            copy16_g2l(&imgb [(size_t)(r0 + srow + 32) * DD + k0 + sc4], &Bs[nxt][srow + 32][sc4]);
        }
        // compute on current buffer (overlaps with async copies)
        #pragma unroll
        for (int kk = 0; kk < 32; kk += 4) {
            const int kb = kk + lk2;
            v2f af;  af.x  = As[cur][mw * 16 + lm][kb];       af.y  = As[cur][mw * 16 + lm][kb + 1];
            v2f bf0; bf0.x = Bs[cur][nw * 32 + lm][kb];       bf0.y = Bs[cur][nw * 32 + lm][kb + 1];
            v2f bf1; bf1.x = Bs[cur][nw * 32 + 16 + lm][kb];  bf1.y = Bs[cur][nw * 32 + 16 + lm][kb + 1];
            acc0 = wmma_f32(af, bf0, acc0);
            acc1 = wmma_f32(af, bf1, acc1);
        }
        copy_wait();       // this wave's async copies for `nxt` have landed
        __syncthreads();   // all waves done reading `cur` and copying `nxt`
    }

    // epilogue: mask, 0 -> NEG, store
    const int n  = lm;
    const int ra = r0 + nw * 32 + n;
    const int rb = ra + 16;
    const float ima = (float)imask[b * NR + ra];
    const float imb = (float)imask[b * NR + rb];
    float* attb = att + (size_t)b * NS * NR;
    #pragma unroll
    for (int i = 0; i < 8; ++i) {
        const int m = i + (hh << 3);
        const int s = s0 + mw * 16 + m;
        const float sm = (float)smask[b * NS + s];
        const float v0 = acc0[i] * (sm * ima);
        const float v1 = acc1[i] * (sm * imb);
        attb[(size_t)s * NR + ra] = (v0 != 0.f) ? v0 : NEGV;
        attb[(size_t)s * NR + rb] = (v1 != 0.f) ? v1 : NEGV;
    }
}

// ---------------------------------------------------------------------------
// Kernel 2: per-span softmax over r, att_first = w.I, partial MSE(span,att_first)
// Block covers 64 span rows of one batch.  (att region is only 4B-aligned in
// d_out, so this tile is loaded with regular b32 loads.)
// ---------------------------------------------------------------------------
__global__ __launch_bounds__(256) void first_kernel(
    const float* __restrict__ span, const float* __restrict__ image,
    const int* __restrict__ smask, const int* __restrict__ imask,
    const float* __restrict__ att, float* __restrict__ att_first,
    float* __restrict__ score) {
    __shared__ float Wf[64][257];   // M=64 x K=256, padded stride
    __shared__ float red[256];

    const int b   = blockIdx.y;
    const int s0  = blockIdx.x * 64;
    const int tid = threadIdx.x;

    const float* attb = att + (size_t)b * NS * NR;
    for (int idx = tid; idx < 64 * NR; idx += 256) {
        const int r = idx & (NR - 1), s = idx >> 8;
        Wf[s][r] = attb[(size_t)(s0 + s) * NR + r];
    }
    __syncthreads();

    // softmax: 16 half-wave processors, 4 rows each
    const int hw = tid >> 4, ln = tid & 15;
    for (int rr = 0; rr < 4; ++rr) {
        const int row = hw * 4 + rr;
        float mx = -3.4e38f;
        for (int c = ln; c < NR; c += 16) mx = fmaxf(mx, Wf[row][c]);
        for (int off = 8; off >= 1; off >>= 1) mx = fmaxf(mx, __shfl_xor(mx, off, 32));
        float sum = 0.f;
        for (int c = ln; c < NR; c += 16) sum += __expf(Wf[row][c] - mx);
        for (int off = 8; off >= 1; off >>= 1) sum += __shfl_xor(sum, off, 32);
        const float inv = 1.f / sum;
        const float sm  = (float)smask[b * NS + s0 + row];
        for (int c = ln; c < NR; c += 16)
            Wf[row][c] = __expf(Wf[row][c] - mx) * inv * sm * (float)imask[b * NR + c];
    }
    __syncthreads();

    // GEMM: (64 x 256) . (256 x 1024) with fused store + MSE
    const int wave = tid >> 5, lane = tid & 31;
    const int lm = lane & 15, lk2 = (lane >> 4) << 1, hh = lane >> 4;
    const float* imgb  = image + (size_t)b * NR * DD;
    const float* spanb = span  + (size_t)b * NS * DD + (size_t)s0 * DD;
    float* outb = att_first + (size_t)b * NS * DD + (size_t)s0 * DD;

    float local = 0.f;
    for (int nt = 0; nt < 8; ++nt) {
        const int n0 = wave * 128 + nt * 16;
        const int nc = n0 + lm;
        v8f acc[4];
        #pragma unroll
        for (int i = 0; i < 4; ++i) acc[i] = zero8();
        for (int k = 0; k < NR; k += 4) {
            const int kb = k + lk2;
            v2f bf; bf.x = imgb[(size_t)kb * DD + nc];
                    bf.y = imgb[(size_t)(kb + 1) * DD + nc];
            #pragma unroll
            for (int ms = 0; ms < 4; ++ms) {
                v2f af; af.x = Wf[ms * 16 + lm][kb];
                        af.y = Wf[ms * 16 + lm][kb + 1];
                acc[ms] = wmma_f32(af, bf, acc[ms]);
            }
        }
        #pragma unroll
        for (int ms = 0; ms < 4; ++ms) {
            #pragma unroll
            for (int i = 0; i < 8; ++i) {
                const int m = ms * 16 + i + (hh << 3);
                const float v = acc[ms][i];
                outb[(size_t)m * DD + nc] = v;
                const float d = spanb[(size_t)m * DD + nc] - v;
                local += d * d;
            }
        }
    }
    red[tid] = local;
    __syncthreads();
    for (int off = 128; off >= 1; off >>= 1) {
        if (tid < off) red[tid] += red[tid + off];
        __syncthreads();
    }
    if (tid == 0) atomicAdd(score, red[0] * (1.f / 33554432.f));  // 1/(B*Ns*D)
}

// ---------------------------------------------------------------------------
// Kernel 3: per-ROI softmax over s, att_second = w.S, partial MSE(image, .)
// Block covers 32 ROI rows of one batch.
// ---------------------------------------------------------------------------
__global__ __launch_bounds__(256) void second_kernel(
    const float* __restrict__ span, const float* __restrict__ image,
    const int* __restrict__ smask, const int* __restrict__ imask,
    const float* __restrict__ att, float* __restrict__ score) {
    __shared__ float Wt[32][513];   // M=32 x K=512, padded stride
    __shared__ float red[256];

    const int b   = blockIdx.y;
    const int r0  = blockIdx.x * 32;
    const int tid = threadIdx.x;

    const float* attb = att + (size_t)b * NS * NR;
    for (int idx = tid; idx < NS * 32; idx += 256) {
        const int rl = idx & 31, s = idx >> 5;
        Wt[rl][s] = attb[(size_t)s * NR + r0 + rl];   // transpose into LDS
    }
    __syncthreads();

    const int hw = tid >> 4, ln = tid & 15;
    for (int rr = 0; rr < 2; ++rr) {
        const int row = hw * 2 + rr;
        float mx = -3.4e38f;
        for (int c = ln; c < NS; c += 16) mx = fmaxf(mx, Wt[row][c]);
        for (int off = 8; off >= 1; off >>= 1) mx = fmaxf(mx, __shfl_xor(mx, off, 32));
        float sum = 0.f;
        for (int c = ln; c < NS; c += 16) sum += __expf(Wt[row][c] - mx);
        for (int off = 8; off >= 1; off >>= 1) sum += __shfl_xor(sum, off, 32);
        const float inv = 1.f / sum;
        const float im  = (float)imask[b * NR + r0 + row];
        for (int c = ln; c < NS; c += 16)
            Wt[row][c] = __expf(Wt[row][c] - mx) * inv * im * (float)smask[b * NS + c];
    }
    __syncthreads();

    // GEMM: (32 x 512) . (512 x 1024), fused MSE only (att_second not an output)
    const int wave = tid >> 5, lane = tid & 31;
    const int lm = lane & 15, lk2 = (lane >> 4) << 1, hh = lane >> 4;
    const float* spanb = span  + (size_t)b * NS * DD;
    const float* imgb  = image + (size_t)b * NR * DD + (size_t)r0 * DD;

    float local = 0.f;
    for (int nt = 0; nt < 8; ++nt) {
        const int n0 = wave * 128 + nt * 16;
        const int nc = n0 + lm;
        v8f acc[2];
        acc[0] = zero8(); acc[1] = zero8();
        for (int k = 0; k < NS; k += 4) {
            const int kb = k + lk2;
            v2f bf; bf.x = spanb[(size_t)kb * DD + nc];
                    bf.y = spanb[(size_t)(kb + 1) * DD + nc];
            #pragma unroll
            for (int ms = 0; ms < 2; ++ms) {
                v2f af; af.x = Wt[ms * 16 + lm][kb];
                        af.y = Wt[ms * 16 + lm][kb + 1];
                acc[ms] = wmma_f32(af, bf, acc[ms]);
            }
        }
        #pragma unroll
        for (int ms = 0; ms < 2; ++ms) {
            #pragma unroll
            for (int i = 0; i < 8; ++i) {
                const int m = ms * 16 + i + (hh << 3);
                const float v = acc[ms][i];
                const float d = imgb[(size_t)m * DD + nc] - v;
                local += d * d;
            }
        }
    }
    red[tid] = local;
    __syncthreads();
    for (int off = 128; off >= 1; off >>= 1) {
        if (tid < off) red[tid] += red[tid + off];
        __syncthreads();
    }
    if (tid == 0) atomicAdd(score, red[0] * (1.f / 16777216.f));  // 1/(B*Nr*D)
}

// ---------------------------------------------------------------------------
// Host launcher
// d_out layout: [score(1)] [att_first (B*Ns*D)] [att (B*Ns*Nr)]
// ---------------------------------------------------------------------------
extern "C" void kernel_launch(void* const* d_in, const int* in_sizes, int n_in,
                              void* d_out, int out_size, void* d_ws, size_t ws_size,
                              hipStream_t stream) {
    const float* span  = (const float*)d_in[0];
    const float* image = (const float*)d_in[1];
    const int*   smask = (const int*)d_in[2];
    const int*   imask = (const int*)d_in[3];

    float* out       = (float*)d_out;
    float* score     = out;
    float* att_first = out + 1;
    float* att       = out + 1 + (size_t)B_ * NS * DD;

    hipLaunchKernelGGL(init_kernel, dim3(1), dim3(1), 0, stream, score);
    hipLaunchKernelGGL(att_kernel, dim3(NS / 64, NR / 64, B_), dim3(256), 0, stream,
                       span, image, smask, imask, att);
    hipLaunchKernelGGL(first_kernel, dim3(NS / 64, B_), dim3(256), 0, stream,
                       span, image, smask, imask, att, att_first, score);
    hipLaunchKernelGGL(second_kernel, dim3(NR / 32, B_), dim3(256), 0, stream,
                       span, image, smask, imask, att, score);
}